// DeepSetLevelEmbedding_26774826123403
// MI455X (gfx1250) — compile-verified
//
#include <hip/hip_runtime.h>
#include <hip/hip_bf16.h>

// DeepSetLevelEmbedding: per-row 32-bin histogram of binned cosine values,
// then log2(count+1) * bin_embs[bin].  HBM-bound (128 MiB read, ~5.8 us floor
// at 23.3 TB/s).  CDNA5 path: async global->LDS streaming (ASYNCcnt-pipelined)
// + conflict-free wave32-privatized LDS histogram (ds_add_u32).

#define ROW_N    8192
#define NUM_BINS 32

__global__ __launch_bounds__(256) void
deepset_hist_kernel(const float* __restrict__ cosine,
                    const float* __restrict__ bin_embs,
                    float* __restrict__ out)
{
    __shared__ float    stage[8 * 1024];       // 32 KiB streaming buffer (4 KiB/wave)
    __shared__ unsigned hist[NUM_BINS * 32];   // [bin][lane] privatized, conflict-free
    __shared__ float    red[NUM_BINS * 8];     // reduction scratch

    const unsigned t = threadIdx.x;
    const unsigned w = t >> 5;                 // wave id 0..7 (wave32)
    const unsigned l = t & 31u;                // lane id

    // zero histogram: 256 threads x 4 counters, vectorized
    *(uint4*)&hist[t * 4u] = make_uint4(0u, 0u, 0u, 0u);

    // per-wave streaming bases
    const unsigned stage_base =
        (unsigned)(size_t)&stage[w * 1024u + l * 4u];   // LDS byte address (low 32 bits)
    const int gbase =
        (int)(((size_t)blockIdx.x * ROW_N + (size_t)w * 1024u + (size_t)l * 4u)
              * sizeof(float));                          // < 2^27, fits signed 32-bit

    // one async instruction moves 512 B (32 lanes x b128) into this wave's staging
#define ISSUE(s)                                                              \
    do {                                                                      \
        unsigned _laddr = stage_base + (unsigned)(s) * 512u;                  \
        int      _goff  = gbase + (s) * 512;                                  \
        asm volatile("global_load_async_to_lds_b128 %0, %1, %2"               \
                     :: "v"(_laddr), "v"(_goff), "s"(cosine) : "memory");     \
    } while (0)

#define WAITA(n) asm volatile("s_wait_asynccnt %0" :: "i"(n) : "memory")

#define PROCESS(s)                                                            \
    do {                                                                      \
        const float4 v = *(const float4*)&stage[w * 1024u + (s) * 128u + l * 4u]; \
        float c0 = fminf(fmaxf(v.x, -0.999f), 0.999f);                        \
        float c1 = fminf(fmaxf(v.y, -0.999f), 0.999f);                        \
        float c2 = fminf(fmaxf(v.z, -0.999f), 0.999f);                        \
        float c3 = fminf(fmaxf(v.w, -0.999f), 0.999f);                        \
        int i0 = (int)floorf(c0 * 16.0f) + 16;                                \
        int i1 = (int)floorf(c1 * 16.0f) + 16;                                \
        int i2 = (int)floorf(c2 * 16.0f) + 16;                                \
        int i3 = (int)floorf(c3 * 16.0f) + 16;                                \
        atomicAdd(&hist[(unsigned)i0 * 32u + l], 1u);                         \
        atomicAdd(&hist[(unsigned)i1 * 32u + l], 1u);                         \
        atomicAdd(&hist[(unsigned)i2 * 32u + l], 1u);                         \
        atomicAdd(&hist[(unsigned)i3 * 32u + l], 1u);                         \
    } while (0)

    // 4-deep async pipeline; async loads complete in order, so a wait of
    // (outstanding-1) guarantees the oldest chunk has landed in LDS.
    ISSUE(0); ISSUE(1); ISSUE(2); ISSUE(3);
    __syncthreads();                      // hist zero visible to all waves
    ISSUE(4); WAITA(3); PROCESS(0);
    ISSUE(5); WAITA(3); PROCESS(1);
    ISSUE(6); WAITA(3); PROCESS(2);
    ISSUE(7); WAITA(3); PROCESS(3);
    WAITA(3); PROCESS(4);
    WAITA(2); PROCESS(5);
    WAITA(1); PROCESS(6);
    WAITA(0); PROCESS(7);

#undef ISSUE
#undef WAITA
#undef PROCESS

    __syncthreads();

    // reduce 1024 counters -> 256 partials (each thread sums 4 via ds_load_b128)
    {
        const unsigned bin = t >> 3, idx = t & 7u;
        const uint4 h = *(const uint4*)&hist[bin * 32u + idx * 4u];
        red[bin * 8u + idx] = (float)(h.x + h.y + h.z + h.w);
    }
    __syncthreads();

    // 256 partials -> 32 counts; apply log2(count+1) * emb[bin]
    if (t < NUM_BINS) {
        const float4 a = *(const float4*)&red[t * 8u];
        const float4 b = *(const float4*)&red[t * 8u + 4u];
        const float cnt = a.x + a.y + a.z + a.w + b.x + b.y + b.z + b.w;
        out[(size_t)blockIdx.x * NUM_BINS + t] = log2f(cnt + 1.0f) * bin_embs[t];
    }
}

extern "C" void kernel_launch(void* const* d_in, const int* in_sizes, int n_in,
                              void* d_out, int out_size, void* d_ws, size_t ws_size,
                              hipStream_t stream) {
    const float* cosine = (const float*)d_in[0];   // [B, 8192] f32
    const float* embs   = (const float*)d_in[1];   // [32, 1]  f32
    float*       out    = (float*)d_out;           // [B, 32]  f32

    const int rows = in_sizes[0] / ROW_N;          // B = 4096
    deepset_hist_kernel<<<dim3(rows), dim3(256), 0, stream>>>(cosine, embs, out);
}